// RoIAlign_28664611733860
// MI455X (gfx1250) — compile-verified
//
#include <hip/hip_runtime.h>

// RoIAlign (PH=7, PW=7, SR=2) for MI455X / gfx1250, wave32.
// Gather-bound: 80MB feature map is L2-resident (192MB L2); limit is L2 gather
// transactions. Hot loop does 2 row-pair loads per sample (x-taps fused).
// Output staged in LDS, streamed out with async LDS->global b128 stores.

#define PH 7
#define PW 7
#define SR 2

constexpr int Cc = 256;
constexpr int Hh = 200;
constexpr int Ww = 200;
constexpr int NS = (PH * SR) * (PW * SR);  // 196 sample points per roi
constexpr int NB = PH * PW;                // 49 output bins per (roi, channel)
constexpr int CH_PER_BLK = 128;            // channels per block
constexpr int THREADS    = 128;            // 4 wave32s
constexpr int CHUNKS     = Cc / CH_PER_BLK;

// 2-float load with truthful 4B alignment: compiler emits global_load_b64 when
// unaligned mode permits, else two b32 loads. Correct either way.
struct __attribute__((packed, aligned(4))) F2 { float x, y; };

__global__ __launch_bounds__(THREADS)
void roialign_kernel(const float* __restrict__ feats,
                     const float* __restrict__ rois,
                     const int*   __restrict__ sidx,
                     float* __restrict__ out)
{
    __shared__ __align__(16) int2   s_off[NS];              // row-pair base offsets
    __shared__ __align__(16) float4 s_wt[NS];               // fused weights
    __shared__ __align__(16) float  s_tile[CH_PER_BLK * NB];// staged output (25 KB)

    const int n     = blockIdx.x / CHUNKS;
    const int chunk = blockIdx.x % CHUNKS;
    const int t     = threadIdx.x;

    // ---- Phase 1: per-roi sample table (shared by all channels) -------------
    {
        const float x1 = rois[n * 4 + 0];
        const float y1 = rois[n * 4 + 1];
        const float x2 = rois[n * 4 + 2];
        const float y2 = rois[n * 4 + 3];
        const float binw = fmaxf(x2 - x1, 1.0f) / (float)PW;
        const float binh = fmaxf(y2 - y1, 1.0f) / (float)PH;

        for (int s = t; s < NS; s += THREADS) {
            const int iy = s / (PW * SR);
            const int ix = s % (PW * SR);
            const float ys = y1 + (((float)iy + 0.5f) / (float)SR) * binh;
            const float xs = x1 + (((float)ix + 0.5f) / (float)SR) * binw;

            const bool vy = (ys >= -1.0f) && (ys <= (float)Hh);
            const bool vx = (xs >= -1.0f) && (xs <= (float)Ww);

            const float yc = fminf(fmaxf(ys, 0.0f), (float)(Hh - 1));
            const float xc = fminf(fmaxf(xs, 0.0f), (float)(Ww - 1));
            const int y0  = (int)floorf(yc);
            const int x0  = (int)floorf(xc);
            const int y1i = min(y0 + 1, Hh - 1);
            const int x1i = min(x0 + 1, Ww - 1);
            const float ly = yc - (float)y0;
            const float lx = xc - (float)x0;
            const float hy = 1.0f - ly;
            const float hx = 1.0f - lx;

            // fold 2x2 subsample averaging (0.25) and validity mask
            const float m = (vy && vx) ? 0.25f : 0.0f;

            // x-taps are adjacent pixels {x0, x0+1}; at the right-edge clamp
            // (x1i==x0, i.e. x0==W-1, lx==0) shift the window left and put the
            // full x-weight on the .y lane so loads never leave the plane.
            int   xp  = x0;
            float wax = hx;   // weight on pair.x
            float wbx = lx;   // weight on pair.y
            if (x1i == x0) { xp = x0 - 1; wax = 0.0f; wbx = 1.0f; }

            s_off[s] = make_int2(y0 * Ww + xp, y1i * Ww + xp);
            s_wt[s]  = make_float4(hy * wax * m, hy * wbx * m,
                                   ly * wax * m, ly * wbx * m);
        }
    }
    __syncthreads();

    // ---- Phase 2: gather + accumulate (thread = channel) --------------------
    {
        const int c = chunk * CH_PER_BLK + t;
        const size_t plane = (size_t)Hh * (size_t)Ww;
        const float* __restrict__ p =
            feats + ((size_t)sidx[n] * (size_t)Cc + (size_t)c) * plane;

        for (int ph = 0; ph < PH; ++ph) {
            for (int pw = 0; pw < PW; ++pw) {
                float acc = 0.0f;
#pragma unroll
                for (int sy = 0; sy < SR; ++sy) {
#pragma unroll
                    for (int sx = 0; sx < SR; ++sx) {
                        const int s = (ph * SR + sy) * (PW * SR) + (pw * SR + sx);
                        const int2   o = s_off[s];   // ds_load_b64 (broadcast)
                        const float4 w = s_wt[s];    // ds_load_b128 (broadcast)
                        const F2 a  = *(const F2*)(p + o.x);  // row y0 pair
                        const F2 b2 = *(const F2*)(p + o.y);  // row y1 pair
                        acc = fmaf(w.x, a.x,  acc);
                        acc = fmaf(w.y, a.y,  acc);
                        acc = fmaf(w.z, b2.x, acc);
                        acc = fmaf(w.w, b2.y, acc);
                    }
                }
                s_tile[t * NB + ph * PW + pw] = acc;  // per-thread rows contiguous
            }
        }
    }
    __syncthreads();

    // ---- Phase 3: async LDS -> global bulk store (CDNA5, ASYNCcnt) ----------
    // out layout: [n][c][ph][pw]; this block owns a contiguous 128*49 slab.
    {
        float* __restrict__ gbase =
            out + ((size_t)n * (size_t)Cc + (size_t)chunk * CH_PER_BLK) * (size_t)NB;
        constexpr int NV4 = (CH_PER_BLK * NB) / 4;  // 1568 x 16B, wave-aligned tail
        for (int i = t; i < NV4; i += THREADS) {
            const float* ga = gbase + (size_t)i * 4;
            unsigned lds_off = (unsigned)(unsigned long long)(const void*)&s_tile[i * 4];
            asm volatile("global_store_async_from_lds_b128 %0, %1, off"
                         :: "v"(ga), "v"(lds_off)
                         : "memory");
        }
#if __has_builtin(__builtin_amdgcn_s_wait_asynccnt)
        __builtin_amdgcn_s_wait_asynccnt(0);
#else
        asm volatile("s_wait_asynccnt 0" ::: "memory");
#endif
    }
}

extern "C" void kernel_launch(void* const* d_in, const int* in_sizes, int n_in,
                              void* d_out, int out_size, void* d_ws, size_t ws_size,
                              hipStream_t stream) {
    (void)n_in; (void)out_size; (void)d_ws; (void)ws_size;
    const float* feats = (const float*)d_in[0];   // (2,256,200,200) f32
    const float* rois  = (const float*)d_in[1];   // (N,4) f32
    const int*   sidx  = (const int*)d_in[2];     // (N,) i32
    const int N = in_sizes[2];                    // 1000
    roialign_kernel<<<dim3(N * CHUNKS), dim3(THREADS), 0, stream>>>(
        feats, rois, sidx, (float*)d_out);
}